// Net_87935160418608
// MI455X (gfx1250) — compile-verified
//
#include <hip/hip_runtime.h>

typedef __attribute__((ext_vector_type(2))) float v2f;
typedef __attribute__((ext_vector_type(8))) float v8f;

#define NNODES  100000
#define NFEAT_  512
#define NHID_   256
#define NOUT_   64
#define NLAYERS 8
#define ALPHA_  0.1f

// ---------------- degree / symmetric normalization ----------------
__global__ void init_deg_kernel(int* deg, int n) {
  int i = blockIdx.x * blockDim.x + threadIdx.x;
  if (i < n) deg[i] = 0;
}

__global__ void deg_accum_kernel(const int* __restrict__ rows, int* deg, int E) {
  int e = blockIdx.x * blockDim.x + threadIdx.x;
  if (e < E) atomicAdd(&deg[rows[e]], 1);
}

__global__ void dinv_kernel(const int* __restrict__ deg, float* dinv, int n) {
  int i = blockIdx.x * blockDim.x + threadIdx.x;
  if (i < n) dinv[i] = rsqrtf((float)deg[i] + 1.0f);  // +1 = self loop
}

// ---------------- weight re-pack: Wp[p][col][{0,1}] = {W[2p][col], W[2p+1][col]}
// makes each lane's WMMA B-fragment one contiguous 8-byte load
__global__ void pack_w_kernel(const float* __restrict__ W, float* __restrict__ Wp,
                              int K, int N) {
  int idx = blockIdx.x * blockDim.x + threadIdx.x;
  int total = (K / 2) * N;
  if (idx >= total) return;
  int p = idx / N;
  int col = idx - p * N;
  Wp[(size_t)idx * 2 + 0] = W[(size_t)(2 * p) * N + col];
  Wp[(size_t)idx * 2 + 1] = W[(size_t)(2 * p + 1) * N + col];
}

// ---------------- SpMM: agg = D^-1/2 (A+I) D^-1/2 @ h ----------------
// self-loop term initializes agg: agg[i] = dinv[i]^2 * h[i]
__global__ __launch_bounds__(NHID_) void agg_init_kernel(
    const float* __restrict__ h, const float* __restrict__ dinv,
    float* __restrict__ agg) {
  int i = blockIdx.x;
  int f = threadIdx.x;
  float d = dinv[i];
  size_t idx = (size_t)i * NHID_ + f;
  agg[idx] = d * d * h[idx];
}

// one wave per edge: coalesced 256-float row gather, native f32 atomic scatter
__global__ __launch_bounds__(256) void spmm_edge_kernel(
    const int* __restrict__ rows, const int* __restrict__ cols,
    const float* __restrict__ dinv, const float* __restrict__ h,
    float* __restrict__ agg, int E) {
  int wave = (blockIdx.x * (int)blockDim.x + (int)threadIdx.x) >> 5;
  int lane = threadIdx.x & 31;
  if (wave >= E) return;
  int r = rows[wave];
  int c = cols[wave];
  float w = dinv[r] * dinv[c];
  const float* __restrict__ src = h + (size_t)c * NHID_;
  float* __restrict__ dst = agg + (size_t)r * NHID_;
#pragma unroll
  for (int i = 0; i < NHID_ / 32; ++i) {
    int f = i * 32 + lane;
    unsafeAtomicAdd(&dst[f], w * src[f]);  // global_atomic_add_f32, no return
  }
}

// ---------------- fused GEMM: C = relu((MIX?0.9A+0.1X0:A) @ W + bias + res) ---
// f32 WMMA 16x16x4, one wave = 16 rows x 64 cols, marching pointers, packed W.
// A frag: lane L -> A[row_base + (L&15)][k + 2*(L>>4) .. +1]  (8B contiguous)
// B frag: lane L -> Wp pair p=k/2+(L>>4), col_base+t*16+(L&15)  (8B contiguous)
// C/D:    vgpr j -> row = row_base + j + 8*(L>>4), col = col_base+t*16+(L&15)
template <bool MIX, bool BIAS, bool RES>
__global__ __launch_bounds__(256) void gemm_wmma_kernel(
    const float* __restrict__ A, const float* __restrict__ X0,
    const float* __restrict__ Wp, const float* __restrict__ bias,
    const float* __restrict__ res, float* __restrict__ C,
    int M, int K, int N) {
  const int lane = threadIdx.x & 31;
  const int wid  = threadIdx.x >> 5;
  const int row_base = (blockIdx.x * 8 + wid) * 16;
  if (row_base >= M) return;  // wave-uniform exit: EXEC all-ones at WMMA
  const int col_base = blockIdx.y * 64;
  const int m  = lane & 15;
  const int kh = lane >> 4;

  v8f acc[4] = {};
  const float* __restrict__ arow = A + (size_t)(row_base + m) * K + 2 * kh;
  const float* __restrict__ xrow =
      MIX ? X0 + (size_t)(row_base + m) * K + 2 * kh : nullptr;
  const float* __restrict__ wp =
      Wp + ((size_t)kh * N + col_base + m) * 2;  // pair kh, this lane's col

  for (int k = 0; k < K; k += 4) {
    v2f a = *(const v2f*)arow;
    arow += 4;
    if (MIX) {  // fused GCNII mix: (1-alpha)*agg + alpha*x0
      v2f x = *(const v2f*)xrow;
      xrow += 4;
      a = a * (1.0f - ALPHA_) + x * ALPHA_;
    }
#pragma unroll
    for (int t = 0; t < 4; ++t) {
      v2f b = *(const v2f*)(wp + t * 32);  // +16 cols -> +32 packed floats
      acc[t] = __builtin_amdgcn_wmma_f32_16x16x4_f32(
          false, a, false, b, (short)0, acc[t], false, false);
    }
    wp += 4 * (size_t)N;  // pair index +2 per k-step
  }

#pragma unroll
  for (int t = 0; t < 4; ++t) {
    const int col = col_base + t * 16 + m;
    const float bv = BIAS ? bias[col] : 0.0f;
#pragma unroll
    for (int j = 0; j < 8; ++j) {
      const int row = row_base + 8 * kh + j;
      float v = acc[t][j] + bv;
      if (RES) v += res[(size_t)row * N + col];
      C[(size_t)row * N + col] = fmaxf(v, 0.0f);
    }
  }
}

// ---------------- driver ----------------
extern "C" void kernel_launch(void* const* d_in, const int* in_sizes, int n_in,
                              void* d_out, int out_size, void* d_ws, size_t ws_size,
                              hipStream_t stream) {
  const float* x    = (const float*)d_in[0];
  const int*   edge = (const int*)d_in[1];  // (2,E) int32, row-major flat
  const float* W1   = (const float*)d_in[2];
  const float* b1   = (const float*)d_in[3];
  const float* W2   = (const float*)d_in[4];
  const float* b2   = (const float*)d_in[5];
  const float* cw   = (const float*)d_in[6];  // (8,256,256)

  const int E = in_sizes[1] / 2;
  const int* rows = edge;
  const int* cols = edge + E;

  const size_t featBytes = (size_t)NNODES * NHID_ * sizeof(float);
  char* ws = (char*)d_ws;
  int*   deg  = (int*)ws;    ws += ((size_t)NNODES * 4 + 255) & ~(size_t)255;
  float* dinv = (float*)ws;  ws += ((size_t)NNODES * 4 + 255) & ~(size_t)255;
  float* wp1  = (float*)ws;  ws += (size_t)NFEAT_ * NHID_ * 4;            // 512KB
  float* wpc  = (float*)ws;  ws += (size_t)NLAYERS * NHID_ * NHID_ * 4;   // 2MB
  float* wp2  = (float*)ws;  ws += (size_t)NHID_ * NOUT_ * 4;             // 64KB
  float* x0   = (float*)ws;  ws += featBytes;
  float* hB   = (float*)ws;  ws += featBytes;
  float* hC   = (float*)ws;  ws += featBytes;
  float* agg  = (float*)ws;  ws += featBytes;

  // gcn_norm
  init_deg_kernel<<<(NNODES + 255) / 256, 256, 0, stream>>>(deg, NNODES);
  deg_accum_kernel<<<(E + 255) / 256, 256, 0, stream>>>(rows, deg, E);
  dinv_kernel<<<(NNODES + 255) / 256, 256, 0, stream>>>(deg, dinv, NNODES);

  // pack all weight matrices into K-pair-interleaved layout (cheap, streamed)
  {
    int t1 = (NFEAT_ / 2) * NHID_;
    pack_w_kernel<<<(t1 + 255) / 256, 256, 0, stream>>>(W1, wp1, NFEAT_, NHID_);
    int tc = (NHID_ / 2) * NHID_;
    for (int l = 0; l < NLAYERS; ++l)
      pack_w_kernel<<<(tc + 255) / 256, 256, 0, stream>>>(
          cw + (size_t)l * NHID_ * NHID_, wpc + (size_t)l * NHID_ * NHID_,
          NHID_, NHID_);
    int t2 = (NHID_ / 2) * NOUT_;
    pack_w_kernel<<<(t2 + 255) / 256, 256, 0, stream>>>(W2, wp2, NHID_, NOUT_);
  }

  const int mBlocks = (NNODES / 16 + 7) / 8;  // 6250 wave-tiles -> 782 blocks

  // x0 = relu(x @ W1 + b1)
  gemm_wmma_kernel<false, true, false>
      <<<dim3(mBlocks, NHID_ / 64), 256, 0, stream>>>(
          x, nullptr, wp1, b1, nullptr, x0, NNODES, NFEAT_, NHID_);

  const float* cur = x0;
  float* bufs[2] = {hB, hC};
  for (int l = 0; l < NLAYERS; ++l) {
    agg_init_kernel<<<NNODES, NHID_, 0, stream>>>(cur, dinv, agg);
    spmm_edge_kernel<<<(E + 7) / 8, 256, 0, stream>>>(rows, cols, dinv, cur,
                                                      agg, E);
    float* nxt = bufs[l & 1];
    gemm_wmma_kernel<true, false, true>
        <<<dim3(mBlocks, NHID_ / 64), 256, 0, stream>>>(
            agg, x0, wpc + (size_t)l * NHID_ * NHID_, nullptr, cur, nxt,
            NNODES, NHID_, NHID_);
    cur = nxt;
  }

  // out = relu(h @ W2 + b2)
  gemm_wmma_kernel<false, true, false>
      <<<dim3(mBlocks, NOUT_ / 64), 256, 0, stream>>>(
          cur, nullptr, wp2, b2, nullptr, (float*)d_out, NNODES, NHID_, NOUT_);
}